// MatryoshkaQuantizer_16587163697890
// MI455X (gfx1250) — compile-verified
//
#include <hip/hip_runtime.h>

// Matryoshka quantizer brute-force argmin, factored into a K=4 GEMM that maps
// exactly onto V_WMMA_F32_16X16X4_F32.
//
//   err(e,q) = [scale^2, -2*y*scale, 1.75*y^2, 0] . [b(q), a(q), 1, 0]
// with y = x + scale*(zero-128), and per-q tables over centered slice codes:
//   a(q) = d8 + 0.5*d4 + 0.25*d2
//   b(q) = d8^2 + 0.5*d4^2 + 0.25*d2^2
//   c(q) = d8 + d4 + d2            (for loss_sel = 3y - scale*c(q*))
//
// The B operand is pre-packed in LDS in the exact WMMA B register layout
// (per tile, per lane), so the hot loop is: ds_load_b64 -> v_wmma -> min/sel,
// with fully uniform EXEC (no divergent selects around the matrix op).

typedef float v2f  __attribute__((ext_vector_type(2)));
typedef float v8f  __attribute__((ext_vector_type(8)));

#define NQ 256
#define NTILES 16
#define ELEMS_PER_WAVE 16
#define WAVES_PER_BLOCK 8
#define ELEMS_PER_BLOCK (ELEMS_PER_WAVE * WAVES_PER_BLOCK)

__device__ __forceinline__ void mq_tables(int q, float& a, float& b, float& c) {
    int d8 = q - 128;
    int s4 = (q >> 4) + ((q >> 3) & 1); if (s4 > 15) s4 = 15;
    int d4 = (s4 << 4) - 128;
    int s2 = (q >> 6) + ((q >> 5) & 1); if (s2 > 3) s2 = 3;
    int d2 = (s2 << 6) - 128;
    a = (float)d8 + 0.5f * (float)d4 + 0.25f * (float)d2;
    b = (float)(d8 * d8) + 0.5f * (float)(d4 * d4) + 0.25f * (float)(d2 * d2);
    c = (float)(d8 + d4 + d2);
}

__global__ __launch_bounds__(256)
void matryoshka_argmin_wmma(const float* __restrict__ x,
                            const float* __restrict__ scale,
                            const float* __restrict__ zero,
                            float* __restrict__ outQ,
                            float* __restrict__ outLoss,
                            int nElem, int cols)
{
    // B operand pre-packed per (tile, lane) in WMMA B layout:
    //   lanes 0-15  (N=lane):  {K0,K1} = {b(q), a(q)},  q = 16*t + lane
    //   lanes 16-31 (N=lane-16): {K2,K3} = {1.0, 0.0}
    __shared__ v2f   Bpack[NTILES * 32];   // 4 KB
    __shared__ float cTab[NQ];             // 1 KB

    // ---- one-time table build (divergence here is harmless) ----
    for (int idx = threadIdx.x; idx < NTILES * 32; idx += 256) {
        const int t = idx >> 5;
        const int L = idx & 31;
        const int q = t * 16 + (L & 15);
        float a, b, c;
        mq_tables(q, a, b, c);
        v2f val;
        if (L < 16) { val.x = b;    val.y = a;    }
        else        { val.x = 1.0f; val.y = 0.0f; }
        Bpack[idx] = val;
    }
    {
        const int q = threadIdx.x;   // blockDim.x == 256 == NQ
        float a, b, c;
        mq_tables(q, a, b, c);
        cTab[q] = c;
    }
    __syncthreads();

    const int lane     = threadIdx.x & 31;
    const int waveId   = threadIdx.x >> 5;
    const int waveBase = blockIdx.x * ELEMS_PER_BLOCK + waveId * ELEMS_PER_WAVE;
    const int n        = lane & 15;       // N position within 16x16 tile
    const bool hi      = lane >= 16;      // upper half-wave: K=2,3 (A) / M+8 (D)

    // Lanes L and L+16 carry the same element m = L&15 (different K slots of A).
    int eA = waveBase + n; if (eA >= nElem) eA = nElem - 1;   // clamp keeps EXEC uniform
    const float xA = x[eA];
    const int   rA = eA / cols;
    const float sA = scale[rA];
    const float zA = zero[rA];
    const float yA = xA + sA * (zA - 128.0f);

    // A (16x4 f32): lanes 0-15 hold K0,K1; lanes 16-31 hold K2,K3 (M = lane&15)
    v2f Aop;
    Aop.x = hi ? (1.75f * yA * yA) : (sA * sA);
    Aop.y = hi ? 0.0f              : (-2.0f * yA * sA);

    float best[8];
    int   bq[8];
#pragma unroll
    for (int v = 0; v < 8; ++v) { best[v] = 3.4e38f; bq[v] = 0; }

    const v8f czero = {};

    // ---- enumerate all 256 codes as 16 q-tiles of 16, one WMMA each ----
#pragma unroll
    for (int t = 0; t < NTILES; ++t) {
        const v2f Bop = Bpack[t * 32 + lane];   // single uniform ds_load_b64
        const v8f d = __builtin_amdgcn_wmma_f32_16x16x4_f32(
            false, Aop, false, Bop, (short)0, czero, false, false);
        // D layout: VGPR v -> element M = v + (hi?8:0), column N = lane&15.
        // Ascending t => ascending q; strict '<' keeps the first minimum
        // (matches jnp.argmin tie-breaking along each lane).
        const int q = t * 16 + n;
#pragma unroll
        for (int v = 0; v < 8; ++v) {
            const float f = d[v];
            if (f < best[v]) { best[v] = f; bq[v] = q; }
        }
    }

    // ---- cross-lane argmin within each 16-lane half (xor masks < 16 never
    //      cross the half boundary on wave32); tie-break toward smaller q ----
#pragma unroll
    for (int v = 0; v < 8; ++v) {
        float bv = best[v];
        int   bi = bq[v];
#pragma unroll
        for (int m = 1; m < 16; m <<= 1) {
            const float ov = __shfl_xor(bv, m, 32);
            const int   oi = __shfl_xor(bi, m, 32);
            if (ov < bv || (ov == bv && oi < bi)) { bv = ov; bi = oi; }
        }
        best[v] = bv; bq[v] = bi;
    }

    // ---- writers: lane with (lane&15)==v owns element m = v + 8*hi ----
    if (n < 8) {
        const int v = n;
        const int e = waveBase + v + (hi ? 8 : 0);
        if (e < nElem) {
            const int   qb = bq[v];
            const int   r  = e / cols;
            const float s  = scale[r];
            const float z  = zero[r];
            const float y  = x[e] + s * (z - 128.0f);
            outQ[e]    = (float)qb;                 // qweight (first output)
            outLoss[e] = 3.0f * y - s * cTab[qb];   // loss_sel (second output)
        }
    }
}

extern "C" void kernel_launch(void* const* d_in, const int* in_sizes, int n_in,
                              void* d_out, int out_size, void* d_ws, size_t ws_size,
                              hipStream_t stream) {
    (void)n_in; (void)out_size; (void)d_ws; (void)ws_size;

    const float* x     = (const float*)d_in[0];
    const float* scale = (const float*)d_in[1];
    const float* zero  = (const float*)d_in[2];
    // d_in[3] = maxq (scalar 255): code enumerates the fixed 256-code space.

    const int nElem = in_sizes[0];            // 1024*256
    const int rows  = in_sizes[1];            // 1024
    const int cols  = nElem / rows;           // 256

    float* outQ    = (float*)d_out;           // qweight, flat
    float* outLoss = (float*)d_out + nElem;   // loss_sel, flat (concatenated)

    const int blocks = (nElem + ELEMS_PER_BLOCK - 1) / ELEMS_PER_BLOCK;
    matryoshka_argmin_wmma<<<blocks, 256, 0, stream>>>(
        x, scale, zero, outQ, outLoss, nElem, cols);
}